// HardQuadTripletSOSRLoss_57982058496723
// MI455X (gfx1250) — compile-verified
//
#include <hip/hip_runtime.h>
#include <math.h>

#define BB 2
#define NN 512
#define CC 128
#define HH 128
#define WWW 128
#define HWHW (HH*WWW)
#define NUMNEG 16
#define SOSNEG 8

typedef __attribute__((ext_vector_type(2))) float v2f;
typedef __attribute__((ext_vector_type(8))) float v8f;

// ---------------------------------------------------------------------------
// 1) Bilinear-sample descriptors at warped keypoints, L2-normalize, compute pos
// ---------------------------------------------------------------------------
__global__ __launch_bounds__(128)
void sample_desc_kernel(const float* __restrict__ desc2,
                        const float* __restrict__ w_kp1,
                        const float* __restrict__ kp1_desc,
                        float* __restrict__ wdesc,
                        float* __restrict__ pos)
{
    const int bn = blockIdx.x;            // b*NN + n
    const int b  = bn / NN;
    const int c  = threadIdx.x;           // channel 0..127

    const float ky = w_kp1[(size_t)bn*2 + 0];
    const float kx = w_kp1[(size_t)bn*2 + 1];
    float y = fminf(fmaxf(ky * 0.125f - 0.5f, 0.0f), (float)(HH-1));
    float x = fminf(fmaxf(kx * 0.125f - 0.5f, 0.0f), (float)(WWW-1));
    int y0 = (int)floorf(y); y0 = min(max(y0, 0), HH-2);
    int x0 = (int)floorf(x); x0 = min(max(x0, 0), WWW-2);
    const float wy = y - (float)y0;
    const float wx = x - (float)x0;

    const float* dp = desc2 + ((size_t)b*CC + c) * HWHW;
    const float v00 = dp[y0*WWW + x0];
    const float v01 = dp[y0*WWW + x0 + 1];
    const float v10 = dp[(y0+1)*WWW + x0];
    const float v11 = dp[(y0+1)*WWW + x0 + 1];
    const float v = v00*(1.f-wy)*(1.f-wx) + v01*(1.f-wy)*wx
                  + v10*wy*(1.f-wx)       + v11*wy*wx;

    __shared__ float red[128];
    red[c] = v*v;
    __syncthreads();
    for (int s = 64; s > 0; s >>= 1) {
        if (c < s) red[c] += red[c+s];
        __syncthreads();
    }
    const float norm = sqrtf(red[0]) + 1e-8f;
    const float outv = v / norm;
    wdesc[(size_t)bn*CC + c] = outv;

    __syncthreads();
    red[c] = outv * kp1_desc[(size_t)bn*CC + c];
    __syncthreads();
    for (int s = 64; s > 0; s >>= 1) {
        if (c < s) red[c] += red[c+s];
        __syncthreads();
    }
    if (c == 0) pos[bn] = 2.0f - 2.0f * red[0];
}

// ---------------------------------------------------------------------------
// Nearest-4 grid cells of a point (matches lax.top_k(-dist,4) incl. tie-break)
// ---------------------------------------------------------------------------
__device__ __forceinline__ void nearest4(float py, float px, int* out4)
{
    int r0 = (int)floorf((py - 4.0f) * 0.125f + 0.5f);
    int c0 = (int)floorf((px - 4.0f) * 0.125f + 0.5f);
    int rlo = min(max(r0 - 2, 0), HH - 5);
    int clo = min(max(c0 - 2, 0), WWW - 5);
    float d2[25];
    int   id[25];
    #pragma unroll
    for (int i = 0; i < 5; ++i) {
        #pragma unroll
        for (int j = 0; j < 5; ++j) {
            const int r = rlo + i, cc = clo + j;
            const float dy = py - (8.0f*r + 4.0f);
            const float dx = px - (8.0f*cc + 4.0f);
            d2[i*5+j] = dy*dy + dx*dx;
            id[i*5+j] = r*WWW + cc;
        }
    }
    #pragma unroll
    for (int s = 0; s < 4; ++s) {
        int best = s;
        for (int t = s+1; t < 25; ++t) {
            if (d2[t] < d2[best] || (d2[t] == d2[best] && id[t] < id[best])) best = t;
        }
        float tv = d2[s]; d2[s] = d2[best]; d2[best] = tv;
        int   ti = id[s]; id[s] = id[best]; id[best] = ti;
        out4[s] = id[s];
    }
}

__global__ void nearest4_points_kernel(const float* __restrict__ pts,
                                       int* __restrict__ ids, int total)
{
    const int i = blockIdx.x * blockDim.x + threadIdx.x;
    if (i >= total) return;
    const float py = pts[(size_t)i*2];
    const float px = pts[(size_t)i*2 + 1];
    int o[4];
    nearest4(py, px, o);
    ids[(size_t)i*4 + 0] = o[0];
    ids[(size_t)i*4 + 1] = o[1];
    ids[(size_t)i*4 + 2] = o[2];
    ids[(size_t)i*4 + 3] = o[3];
}

// Warp grid-cell centers by homography, then nearest-4 of warped points
__global__ void warp_cells_nearest4_kernel(const int* __restrict__ cell_ids,
                                           const float* __restrict__ homo,
                                           int* __restrict__ out_ids)
{
    const int i = blockIdx.x * blockDim.x + threadIdx.x;
    if (i >= BB*4*NN) return;
    const int b  = i / (4*NN);
    const int id = cell_ids[i];
    const float cy = 8.0f * (float)(id / WWW) + 4.0f;
    const float cx = 8.0f * (float)(id % WWW) + 4.0f;
    const float* H = homo + (size_t)b*9;
    const float u = H[0]*cx + H[1]*cy + H[2];
    const float v = H[3]*cx + H[4]*cy + H[5];
    const float w = H[6]*cx + H[7]*cy + H[8];
    const float inv = 1.0f / (w + 1e-8f);
    const float wy = v * inv;
    const float wx = u * inv;
    int o[4];
    nearest4(wy, wx, o);
    out_ids[(size_t)i*4 + 0] = o[0];
    out_ids[(size_t)i*4 + 1] = o[1];
    out_ids[(size_t)i*4 + 2] = o[2];
    out_ids[(size_t)i*4 + 3] = o[3];
}

// ---------------------------------------------------------------------------
// Fused (2 - 2*A.B^T + 5*mask) + row top-K via f32 WMMA (16x16x4).
// MODE 0: dsim (cols = raw desc2, channel-strided), K=16, neigh penalty -> fos partials
// MODE 1: k_sim (cols = kp1_desc), K=8, 4x4 id-match penalty -> ids
// MODE 2: w_sim (cols = wdesc),    K=8, 16x4 id-match penalty -> ids
// ---------------------------------------------------------------------------
template<int MODE>
__global__ __launch_bounds__(256)
void simtopk_kernel(const float* __restrict__ rowdesc,
                    const float* __restrict__ coldesc,
                    const int*   __restrict__ row_ids,
                    const int*   __restrict__ col_ids,
                    const float* __restrict__ pos,
                    float* __restrict__ fosp,
                    int*   __restrict__ outids)
{
    constexpr int KSEL = (MODE == 0) ? NUMNEG : SOSNEG;
    constexpr int NCOL = (MODE == 0) ? HWHW : NN;
    constexpr int NRID = (MODE == 1) ? 4 : 16;

    const int b        = blockIdx.x / (NN/16);
    const int row_base = (blockIdx.x % (NN/16)) * 16;
    const int tid  = threadIdx.x;
    const int wave = tid >> 5;
    const int lane = tid & 31;
    const int lm   = lane & 15;
    const int hi   = lane >> 4;

    __shared__ float s[16][136];
    __shared__ int   cid[128*4];

    // A fragments (16 rows x K=128), resident for the whole kernel.
    // f32 A 16x4 layout: lanes 0-15 hold (K=k0,k0+1), lanes 16-31 hold (K=k0+2,k0+3).
    const float* arow = rowdesc + ((size_t)(b*NN + row_base + lm)) * CC;
    v2f afrag[32];
    #pragma unroll
    for (int f = 0; f < 32; ++f)
        afrag[f] = *(const v2f*)(arow + 4*f + 2*hi);

    float tv[KSEL];
    int   tix[KSEL];
    int   rid[NRID];
    if (tid < 16) {
        #pragma unroll
        for (int k = 0; k < KSEL; ++k) { tv[k] = 3.0e38f; tix[k] = 0; }
        const int row = row_base + tid;
        if (MODE == 1) {
            #pragma unroll
            for (int p = 0; p < NRID; ++p)
                rid[p] = row_ids[(size_t)(b*NN + row)*4 + p];
        } else {
            #pragma unroll
            for (int p = 0; p < NRID; ++p)
                rid[p] = row_ids[((size_t)b*4*NN + (size_t)row*4)*4 + p];
        }
    }

    for (int cb = 0; cb < NCOL; cb += 128) {
        const int col = cb + wave*16;
        v8f acc = {};
        if (MODE == 0) {
            // coldesc = desc2 (B,C,H,W): element (b, k, j) at b*C*HW + k*HW + j
            const float* cp = coldesc + (size_t)b*CC*HWHW + col + lm;
            #pragma unroll
            for (int f = 0; f < 32; ++f) {
                const int k = 4*f + 2*hi;
                v2f bfrag;
                bfrag.x = cp[(size_t)k * HWHW];
                bfrag.y = cp[(size_t)(k+1) * HWHW];
                acc = __builtin_amdgcn_wmma_f32_16x16x4_f32(
                        false, afrag[f], false, bfrag, (short)0, acc, false, false);
            }
        } else {
            const float* cp = coldesc + ((size_t)(b*NN + col + lm)) * CC;
            #pragma unroll
            for (int f = 0; f < 32; ++f) {
                v2f bfrag = *(const v2f*)(cp + 4*f + 2*hi);
                acc = __builtin_amdgcn_wmma_f32_16x16x4_f32(
                        false, afrag[f], false, bfrag, (short)0, acc, false, false);
            }
        }
        // C/D layout: VGPR r, lanes0-15 -> M=r, lanes16-31 -> M=8+r; N=lm
        #pragma unroll
        for (int r = 0; r < 8; ++r)
            s[r + 8*hi][wave*16 + lm] = 2.0f - 2.0f * acc[r];

        if (MODE != 0) {
            // 128 cols x 4 ids = 512 contiguous ints this chunk
            const int* src = col_ids + (size_t)(b*NN + cb)*4;
            cid[2*tid]     = src[2*tid];
            cid[2*tid + 1] = src[2*tid + 1];
        }
        __syncthreads();

        if (tid < 16) {
            float* srow = s[tid];
            if (MODE == 0) {
                #pragma unroll
                for (int t = 0; t < 16; ++t) {
                    const int j = rid[t] - cb;
                    if ((unsigned)j < 128u) srow[j] += 5.0f;
                }
            }
            for (int j = 0; j < 128; ++j) {
                float v = srow[j];
                if (MODE != 0) {
                    const int* cj = &cid[j*4];
                    int cnt = 0;
                    #pragma unroll
                    for (int p = 0; p < NRID; ++p) {
                        const int rp = rid[p];
                        cnt += (rp == cj[0]) + (rp == cj[1]) + (rp == cj[2]) + (rp == cj[3]);
                    }
                    v += 5.0f * (float)cnt;
                }
                if (v < tv[KSEL-1]) {        // strict < keeps top_k's stable tie-break
                    int p = KSEL - 1;
                    while (p > 0 && v < tv[p-1]) {
                        tv[p] = tv[p-1]; tix[p] = tix[p-1]; --p;
                    }
                    tv[p] = v; tix[p] = cb + j;
                }
            }
        }
        __syncthreads();
    }

    if (tid < 16) {
        const int row = row_base + tid;
        if (MODE == 0) {
            const float p = pos[b*NN + row];
            float acc = 0.0f;
            #pragma unroll
            for (int k = 0; k < KSEL; ++k) {
                float r = p - tv[k] + 1.0f;   // MARGIN = 1.0
                r = fmaxf(r, 0.0f);
                acc += r*r;
            }
            fosp[b*NN + row] = acc;
        } else {
            #pragma unroll
            for (int k = 0; k < KSEL; ++k)
                outids[(size_t)(b*NN + row)*SOSNEG + k] = tix[k];
        }
    }
}

// ---------------------------------------------------------------------------
// SOS term: one wave per keypoint
// ---------------------------------------------------------------------------
__global__ __launch_bounds__(32)
void sos_kernel(const float* __restrict__ kdesc,
                const float* __restrict__ wdesc,
                const int*   __restrict__ kids,
                const int*   __restrict__ wids,
                float* __restrict__ sosp)
{
    const int bn   = blockIdx.x;
    const int b    = bn / NN;
    const int lane = threadIdx.x;   // 0..31, 4 channels each

    const float4 kav = *(const float4*)(kdesc + (size_t)bn*CC + lane*4);
    const float4 wav = *(const float4*)(wdesc + (size_t)bn*CC + lane*4);

    float ssum = 0.0f;
    #pragma unroll
    for (int s = 0; s < SOSNEG; ++s) {
        const int kj = kids[(size_t)bn*SOSNEG + s];
        const int wj = wids[(size_t)bn*SOSNEG + s];
        const float4 kbv = *(const float4*)(kdesc + ((size_t)(b*NN + kj))*CC + lane*4);
        const float4 wbv = *(const float4*)(wdesc + ((size_t)(b*NN + wj))*CC + lane*4);
        const float dk = kav.x*kbv.x + kav.y*kbv.y + kav.z*kbv.z + kav.w*kbv.w;
        const float dw = wav.x*wbv.x + wav.y*wbv.y + wav.z*wbv.z + wav.w*wbv.w;
        float part = dw - dk;   // (2-2dk) - (2-2dw) = 2*(dw-dk)
        for (int off = 16; off > 0; off >>= 1)
            part += __shfl_xor(part, off, 32);
        const float vd = 2.0f * part;
        ssum += vd * vd;
    }
    if (lane == 0) sosp[bn] = sqrtf(ssum);
}

// ---------------------------------------------------------------------------
// Final deterministic reduction: loss = mean(fos elems) + mean(sos rows)
// ---------------------------------------------------------------------------
__global__ __launch_bounds__(256)
void final_reduce_kernel(const float* __restrict__ fosp,
                         const float* __restrict__ sosp,
                         float* __restrict__ out)
{
    __shared__ float red[256];
    const int t = threadIdx.x;
    float accF = 0.0f, accS = 0.0f;
    for (int i = t; i < BB*NN; i += 256) { accF += fosp[i]; accS += sosp[i]; }

    red[t] = accF;
    __syncthreads();
    for (int s = 128; s > 0; s >>= 1) {
        if (t < s) red[t] += red[t+s];
        __syncthreads();
    }
    const float fsum = red[0];
    __syncthreads();

    red[t] = accS;
    __syncthreads();
    for (int s = 128; s > 0; s >>= 1) {
        if (t < s) red[t] += red[t+s];
        __syncthreads();
    }
    if (t == 0)
        out[0] = fsum / (float)(BB*NN*NUMNEG) + red[0] / (float)(BB*NN);
}

// ---------------------------------------------------------------------------
extern "C" void kernel_launch(void* const* d_in, const int* in_sizes, int n_in,
                              void* d_out, int out_size, void* d_ws, size_t ws_size,
                              hipStream_t stream)
{
    (void)in_sizes; (void)n_in; (void)out_size; (void)ws_size;
    const float* kp1      = (const float*)d_in[0];
    const float* w_kp1    = (const float*)d_in[1];
    const float* kp1_desc = (const float*)d_in[2];
    const float* desc2    = (const float*)d_in[3];
    const float* homo12   = (const float*)d_in[4];
    float* out = (float*)d_out;

    char* base = (char*)d_ws;
    float* wdesc = (float*)(base + 0);        // B*N*C floats      = 524288 B
    float* pos   = (float*)(base + 524288);   // B*N floats        =   4096 B
    float* fosp  = (float*)(base + 528384);   // B*N floats        =   4096 B
    float* sosp  = (float*)(base + 532480);   // B*N floats        =   4096 B
    int*   kcid  = (int*)  (base + 536576);   // kp1_cell_ids B*N*4 = 16384 B
    int*   wid   = (int*)  (base + 552960);   // w_ids B*N*4        = 16384 B
    int*   kwcc  = (int*)  (base + 569344);   // kp1_w_cell_cell_ids B*4N*4 = 65536 B
    int*   knids = (int*)  (base + 634880);   // B*N*8 ints         = 32768 B
    int*   wnids = (int*)  (base + 667648);   // B*N*8 ints         = 32768 B

    sample_desc_kernel<<<BB*NN, CC, 0, stream>>>(desc2, w_kp1, kp1_desc, wdesc, pos);
    nearest4_points_kernel<<<(BB*NN + 255)/256, 256, 0, stream>>>(kp1, kcid, BB*NN);
    nearest4_points_kernel<<<(BB*NN + 255)/256, 256, 0, stream>>>(w_kp1, wid, BB*NN);
    warp_cells_nearest4_kernel<<<(BB*4*NN + 255)/256, 256, 0, stream>>>(kcid, homo12, kwcc);

    simtopk_kernel<0><<<BB*(NN/16), 256, 0, stream>>>(kp1_desc, desc2, kwcc, nullptr, pos, fosp, nullptr);
    simtopk_kernel<1><<<BB*(NN/16), 256, 0, stream>>>(kp1_desc, kp1_desc, kcid, kcid, nullptr, nullptr, knids);
    simtopk_kernel<2><<<BB*(NN/16), 256, 0, stream>>>(wdesc, wdesc, kwcc, wid, nullptr, nullptr, wnids);

    sos_kernel<<<BB*NN, 32, 0, stream>>>(kp1_desc, wdesc, knids, wnids, sosp);
    final_reduce_kernel<<<1, 256, 0, stream>>>(fosp, sosp, out);
}